// CustomMBGCN_69733089017908
// MI455X (gfx1250) — compile-verified
//
#include <hip/hip_runtime.h>
#include <math.h>

// Problem constants (from reference): G=4, N=64, M=8 -> S=4096, K=64, OUT=128
#define G_ 4
#define N_ 64
#define S_ 4096
#define K_ 64
#define O_ 128

#define SPLIT 16  // S-dimension split for the fused B kernel (grid parallelism)

#define XS 68    // padded LDS stride for 64-float rows (bank-conflict-free strided reads)
#define WS 68
#define TS 68    // padded stride for transposed tiles (XT[64][.], YT/BT[128][.])

typedef __attribute__((ext_vector_type(2))) float v2f;
typedef __attribute__((ext_vector_type(8))) float v8f;

// D = A(16x4, f32) * B(4x16, f32) + C(16x16, f32)  -- CDNA5 V_WMMA_F32_16X16X4_F32
__device__ __forceinline__ v8f wmma4(v2f a, v2f b, v8f c) {
  return __builtin_amdgcn_wmma_f32_16x16x4_f32(
      /*neg_a=*/false, a, /*neg_b=*/false, b,
      /*c_mod=*/(short)0, c, /*reuse_a=*/false, /*reuse_b=*/false);
}

// ---------------------------------------------------------------------------
// K1: per-row inverse L2 norm.  16384 rows, 1 thread/row, float4 loads.
// ---------------------------------------------------------------------------
__global__ void k_rownorm(const float* __restrict__ X, float* __restrict__ inv) {
  int r = blockIdx.x * blockDim.x + threadIdx.x;          // 0 .. G_*S_-1
  const float4* p = (const float4*)(X + (size_t)r * K_);
  float ss = 0.f;
#pragma unroll
  for (int i = 0; i < K_ / 4; ++i) {
    float4 q = p[i];
    ss += q.x * q.x + q.y * q.y + q.z * q.z + q.w * q.w;
  }
  inv[r] = 1.0f / fmaxf(sqrtf(ss), 1e-12f);               // matches clamp_min(norm,1e-12)
}

// ---------------------------------------------------------------------------
// K2: v[g][k] = sum_s X[g][s][k] * inv[g][s]   (column sums of Xn)
// One block per group; 1024 threads = (16 partial slices) x (64 k-lanes).
// ---------------------------------------------------------------------------
__global__ __launch_bounds__(1024) void k_colsum(const float* __restrict__ X,
                                                 const float* __restrict__ inv,
                                                 float* __restrict__ v) {
  __shared__ float part[16][K_];
  int g = blockIdx.x;
  int k = threadIdx.x & (K_ - 1);
  int p = threadIdx.x >> 6;
  const float* Xg = X + (size_t)g * S_ * K_;
  const float* ig = inv + g * S_;
  float acc = 0.f;
  int s0 = p * (S_ / 16), s1 = s0 + (S_ / 16);
  for (int s = s0; s < s1; ++s)
    acc += Xg[(size_t)s * K_ + k] * ig[s];
  part[p][k] = acc;
  __syncthreads();
  if (p == 0) {
    float t = 0.f;
#pragma unroll
    for (int i = 0; i < 16; ++i) t += part[i][k];
    v[g * K_ + k] = t;
  }
}

// ---------------------------------------------------------------------------
// K3: c[g][s] = inv[s] * (deg[s]+1e-8)^(-1/2),  deg[s] = inv[s]*(X[s]·v[g])
// ---------------------------------------------------------------------------
__global__ void k_coef(const float* __restrict__ X, const float* __restrict__ inv,
                       const float* __restrict__ v, float* __restrict__ cf) {
  __shared__ float vs[K_];
  int g = blockIdx.x >> 4;                                 // 16 blocks of 256 rows per group
  if (threadIdx.x < K_) vs[threadIdx.x] = v[g * K_ + threadIdx.x];
  __syncthreads();
  int r = blockIdx.x * blockDim.x + threadIdx.x;
  const float4* x = (const float4*)(X + (size_t)r * K_);
  float dot = 0.f;
#pragma unroll
  for (int i = 0; i < K_ / 4; ++i) {
    float4 q = x[i];
    dot += q.x * vs[4 * i] + q.y * vs[4 * i + 1] + q.z * vs[4 * i + 2] + q.w * vs[4 * i + 3];
  }
  float in  = inv[r];
  float deg = in * dot;
  float dis = 1.0f / sqrtf(deg + 1e-8f);
  cf[r] = in * dis;
}

// ---------------------------------------------------------------------------
// K4: fused double GEMM (the heavy kernel).
//   Per 64-row chunk of t:  Y = diag(c) * (X_chunk @ W^T)   [64 x 128]
//                           Bpart += X_chunk^T @ Y          [64 x 128]
// grid = (G_, SPLIT), block = 1024 (32 waves = 4 x 8 tile grid).
// X is staged both row-major (phase-1 A) and transposed (phase-2 A); the
// scaled Y is stored transposed (YT[o][t]) so the epilogue is 2 float4 LDS
// stores per lane AND phase-2 B fragments are contiguous ds_load_b64.
// c-row values are loaded BEFORE phase 1 so they overlap the WMMA loop.
// ---------------------------------------------------------------------------
__global__ __launch_bounds__(1024) void k_fused_B(const float* __restrict__ X,
                                                  const float* __restrict__ W,
                                                  const float* __restrict__ cf,
                                                  float* __restrict__ Bpart) {
  extern __shared__ float smem[];
  float* ldsW  = smem;                     // 128 x WS
  float* ldsX  = ldsW + O_ * WS;           // 64 x XS   (row-major chunk)
  float* ldsXT = ldsX + 64 * XS;           // 64 x TS   (transposed chunk: XT[k][t])
  float* ldsYT = ldsXT + 64 * TS;          // 128 x TS  (transposed Y: YT[o][t])
  float* ldsC  = ldsYT + O_ * TS;          // 64

  const int g     = blockIdx.x;
  const int chunk = blockIdx.y;            // SPLIT chunks, (64/SPLIT) inner iters each
  const int tid   = threadIdx.x;
  const int lane  = tid & 31, wave = tid >> 5;
  const int half  = lane >> 4, l16 = lane & 15;
  const int ttile = wave >> 3;             // t-tile for Y phase, k-tile for B phase
  const int otile = wave & 7;

  const float* Xg = X + (size_t)g * S_ * K_;
  const float* cg = cf + (size_t)g * S_;

  // Stage W (128x64) into LDS once: 8192 floats = 2 float4 / thread.
  {
    const float4* Wv = (const float4*)W;
#pragma unroll
    for (int i = 0; i < 2; ++i) {
      int f = tid * 2 + i;                 // float4 index over [128][16]
      int row = f >> 4, c4 = f & 15;
      *(float4*)(ldsW + row * WS + c4 * 4) = Wv[f];
    }
  }

  // Per-wave fragment base pointers (constant across iterations).
  const float* ax  = ldsX  + (ttile * 16 + l16) * XS + half * 2;  // phase-1 A
  const float* bw  = ldsW  + (otile * 16 + l16) * WS + half * 2;  // phase-1 B
  const float* axt = ldsXT + (ttile * 16 + l16) * TS + half * 2;  // phase-2 A
  const float* byt = ldsYT + (otile * 16 + l16) * TS + half * 2;  // phase-2 B
  const float* cvp = ldsC + ttile * 16 + half * 8;
  float* ytw = ldsYT + (otile * 16 + l16) * TS + ttile * 16 + half * 8;

  v8f accB = {};                           // persistent 16x16 B-tile accumulator
  const int it0 = chunk * (64 / SPLIT);
  for (int it = it0; it < it0 + (64 / SPLIT); ++it) {
    const int t0 = it * 64;
    __syncthreads();                       // prior-iter LDS reads done before overwrite

    // Stage X chunk (64x64): one float4 per thread, row-major + transposed copy.
    {
      int row = tid >> 4, c4 = tid & 15;
      float4 q = *(const float4*)(Xg + (size_t)(t0 + row) * K_ + c4 * 4);
      *(float4*)(ldsX + row * XS + c4 * 4) = q;
      ldsXT[(c4 * 4 + 0) * TS + row] = q.x;
      ldsXT[(c4 * 4 + 1) * TS + row] = q.y;
      ldsXT[(c4 * 4 + 2) * TS + row] = q.z;
      ldsXT[(c4 * 4 + 3) * TS + row] = q.w;
    }
    if (tid < 64) ldsC[tid] = cg[t0 + tid];
    __syncthreads();

    // Load this wave's 8 c-row values NOW so they overlap the WMMA loop.
    float cv[8];
    *(float4*)(cv)     = *(const float4*)(cvp);
    *(float4*)(cv + 4) = *(const float4*)(cvp + 4);

    // Phase 1: Y_tile[16x16] = X_chunk(16 x 64) @ W^T(64 x 16).
    v8f accY = {};
#pragma unroll
    for (int kb = 0; kb < K_; kb += 32) {  // batches: many b64 loads in flight, 8 wmma
      v2f a[8], b[8];
#pragma unroll
      for (int i = 0; i < 8; ++i) {
        int k = kb + i * 4;
        a[i] = *(const v2f*)(ax + k);      // contiguous pair -> ds_load_b64
        b[i] = *(const v2f*)(bw + k);
      }
#pragma unroll
      for (int i = 0; i < 8; ++i) accY = wmma4(a[i], b[i], accY);
    }

    // Scale rows by c and store TRANSPOSED: YT[o][t]; lane's 8 elements are
    // consecutive t -> two float4 stores, no scattered b32.
    {
      float y0 = accY[0] * cv[0], y1 = accY[1] * cv[1];
      float y2 = accY[2] * cv[2], y3 = accY[3] * cv[3];
      float y4 = accY[4] * cv[4], y5 = accY[5] * cv[5];
      float y6 = accY[6] * cv[6], y7 = accY[7] * cv[7];
      *(float4*)(ytw)     = (float4){y0, y1, y2, y3};
      *(float4*)(ytw + 4) = (float4){y4, y5, y6, y7};
    }
    __syncthreads();

    // Phase 2: B_tile += X_chunk^T(16 x 64) @ Y(64 x 16); both fragment
    // streams are contiguous ds_load_b64 from the transposed copies.
#pragma unroll
    for (int tb = 0; tb < 64; tb += 32) {
      v2f a[8], b[8];
#pragma unroll
      for (int i = 0; i < 8; ++i) {
        int t = tb + i * 4;
        a[i] = *(const v2f*)(axt + t);
        b[i] = *(const v2f*)(byt + t);
      }
#pragma unroll
      for (int i = 0; i < 8; ++i) accB = wmma4(a[i], b[i], accB);
    }
  }

  // Write this chunk's partial B tile: Bpart[(g*SPLIT+chunk)][64][128]
#pragma unroll
  for (int r = 0; r < 8; ++r) {
    int krow = ttile * 16 + half * 8 + r;
    Bpart[(((size_t)(g * SPLIT + chunk)) * K_ + krow) * O_ + otile * 16 + l16] = accB[r];
  }
}

// ---------------------------------------------------------------------------
// K5: reduce the SPLIT partial B buffers per group.
// ---------------------------------------------------------------------------
__global__ void k_reduceB(const float* __restrict__ Bpart, float* __restrict__ Bws) {
  int i = blockIdx.x * blockDim.x + threadIdx.x;   // 0 .. 4*8192-1
  int g = i >> 13;
  int j = i & 8191;
  float s = 0.f;
#pragma unroll
  for (int c = 0; c < SPLIT; ++c) s += Bpart[(((size_t)(g * SPLIT + c)) << 13) + j];
  Bws[i] = s;
}

// ---------------------------------------------------------------------------
// K6: Out[g][n][o] = c[s_n] * X[s_n] @ B[g],  s_n = n*64 + 36 (center pixel).
// One block per group, 32 waves = 4 n-tiles x 8 o-tiles.  B is staged
// transposed so both fragment streams are contiguous ds_load_b64.
// ---------------------------------------------------------------------------
__global__ __launch_bounds__(1024) void k_out(const float* __restrict__ X,
                                              const float* __restrict__ Bws,
                                              const float* __restrict__ cf,
                                              float* __restrict__ Out) {
  extern __shared__ float smem[];
  float* ldsXc = smem;                     // 64 x XS
  float* ldsBT = ldsXc + 64 * XS;          // 128 x TS  (BT[o][k])
  float* ldsC  = ldsBT + O_ * TS;          // 64

  const int g   = blockIdx.x;
  const int tid = threadIdx.x;
  const int lane = tid & 31, wave = tid >> 5;
  const int half = lane >> 4, l16 = lane & 15;
  const int ntile = wave >> 3, otile = wave & 7;

  {                                        // gather the 64 center rows of X
    int row = tid >> 4, c4 = tid & 15;
    size_t s = (size_t)g * S_ + (size_t)row * 64 + 36;
    *(float4*)(ldsXc + row * XS + c4 * 4) = *(const float4*)(X + s * K_ + c4 * 4);
  }
  {                                        // stage B (64x128) transposed: BT[o][k]
#pragma unroll
    for (int i = 0; i < 2; ++i) {
      int f = tid * 2 + i;
      int row = f >> 5, c4 = f & 31;       // row = k, c4 = o-quad
      float4 q = *(const float4*)(Bws + ((size_t)g * K_ + row) * O_ + c4 * 4);
      ldsBT[(c4 * 4 + 0) * TS + row] = q.x;
      ldsBT[(c4 * 4 + 1) * TS + row] = q.y;
      ldsBT[(c4 * 4 + 2) * TS + row] = q.z;
      ldsBT[(c4 * 4 + 3) * TS + row] = q.w;
    }
  }
  if (tid < 64) ldsC[tid] = cf[(size_t)g * S_ + (size_t)tid * 64 + 36];
  __syncthreads();

  // Load scale values early so they overlap the WMMA loop.
  float cv[8];
  *(float4*)(cv)     = *(const float4*)(ldsC + ntile * 16 + half * 8);
  *(float4*)(cv + 4) = *(const float4*)(ldsC + ntile * 16 + half * 8 + 4);

  v8f acc = {};
  const float* ax = ldsXc + (ntile * 16 + l16) * XS + half * 2;
  const float* bb = ldsBT + (otile * 16 + l16) * TS + half * 2;
#pragma unroll
  for (int kb = 0; kb < K_; kb += 32) {
    v2f a[8], b[8];
#pragma unroll
    for (int i = 0; i < 8; ++i) {
      int k = kb + i * 4;
      a[i] = *(const v2f*)(ax + k);
      b[i] = *(const v2f*)(bb + k);
    }
#pragma unroll
    for (int i = 0; i < 8; ++i) acc = wmma4(a[i], b[i], acc);
  }
#pragma unroll
  for (int r = 0; r < 8; ++r) {
    int n = ntile * 16 + half * 8 + r;
    Out[((size_t)g * N_ + n) * O_ + otile * 16 + l16] = acc[r] * cv[r];
  }
}

// ---------------------------------------------------------------------------
extern "C" void kernel_launch(void* const* d_in, const int* in_sizes, int n_in,
                              void* d_out, int out_size, void* d_ws, size_t ws_size,
                              hipStream_t stream) {
  (void)in_sizes; (void)n_in; (void)out_size; (void)ws_size;
  const float* X = (const float*)d_in[0];   // [G,N,M,M,K] contiguous = [G,S,K]
  const float* W = (const float*)d_in[1];   // [OUT,K]
  float* out = (float*)d_out;               // [G,N,OUT] f32

  float* ws    = (float*)d_ws;
  float* inv   = ws;                        // 16384
  float* cf    = ws + 16384;                // 16384
  float* v     = ws + 32768;                //   256
  float* Bws   = ws + 33024;                // 32768
  float* Bpart = ws + 65792;                // SPLIT*4*8192 = 524288 (~2.4 MB total)

  k_rownorm<<<dim3(64), dim3(256), 0, stream>>>(X, inv);
  k_colsum <<<dim3(G_), dim3(1024), 0, stream>>>(X, inv, v);
  k_coef   <<<dim3(64), dim3(256), 0, stream>>>(X, inv, v, cf);

  size_t lds4 = (size_t)(O_ * WS + 64 * XS + 64 * TS + O_ * TS + 64) * sizeof(float);
  k_fused_B<<<dim3(G_, SPLIT), dim3(1024), lds4, stream>>>(X, W, cf, Bpart);

  k_reduceB<<<dim3(128), dim3(256), 0, stream>>>(Bpart, Bws);

  size_t lds6 = (size_t)(64 * XS + O_ * TS + 64) * sizeof(float);
  k_out<<<dim3(G_), dim3(1024), lds6, stream>>>(X, Bws, cf, out);
}